// autoencoder_8693013807607
// MI455X (gfx1250) — compile-verified
//
#include <hip/hip_runtime.h>

// CDNA5 / gfx1250, wave32. One wave handles one 16-row tile of the (4096,16)
// bits matrix; both encoder layers run as V_WMMA_F32_16X16X4_F32 chunks.
// Weights are staged once per block into zero-padded LDS images so the
// B-matrix fragment reads are branch-free ds_loads (no exec-masked vmem).

typedef float v2f __attribute__((ext_vector_type(2)));
typedef float v8f __attribute__((ext_vector_type(8)));

#define WAVES_PER_BLOCK 8
#define ROWS_PER_WAVE   16

__global__ __launch_bounds__(256) void ae_wmma_kernel(
    const float* __restrict__ x,       // (4096,) flattened (256,16)
    const float* __restrict__ enc_w1,  // (8,16) row-major
    const float* __restrict__ enc_b1,  // (8,)
    const float* __restrict__ enc_w2,  // (4,8)
    const float* __restrict__ enc_b2,  // (4,)
    const float* __restrict__ dec_w1,  // (8,16)
    const float* __restrict__ dec_b1,  // (8,)
    const float* __restrict__ dec_w2,  // (4,8)
    const float* __restrict__ dec_b2,  // (4,)
    float* __restrict__ out, int n)
{
  // Zero-padded weight images shared by all 8 waves in the block.
  __shared__ float s_w1t[16][16];  // [k][ncol]: enc_w1^T, cols 8..15 = 0
  __shared__ float s_w2t[8][16];   // [k][ncol]: enc_w2^T, cols 4..15 = 0
  __shared__ float s_b1[16];       // enc_b1 padded with zeros
  __shared__ float s_b2[16];       // enc_b2 padded with zeros
  __shared__ float s_xb[4];        // decoder-on-basis vector (block-wide)
  // Per-wave scratch: h1 tile (stride 17 vs bank conflicts) + final codes.
  __shared__ float lds_h1[WAVES_PER_BLOCK][16][17];
  __shared__ float lds_code[WAVES_PER_BLOCK][16][4];

  const int t = threadIdx.x;

  // ---- Cooperative, branch-light staging (once per block) ----
  {
    const int k  = t >> 4;   // 0..15
    const int nn = t & 15;   // 0..15
    s_w1t[k][nn] = (nn < 8) ? enc_w1[nn * 16 + k] : 0.0f;   // 256 threads
    if (t < 128) s_w2t[k][nn] = (nn < 4) ? enc_w2[nn * 8 + k] : 0.0f;
    if (t < 16)  s_b1[t] = (t < 8) ? enc_b1[t] : 0.0f;
    if (t >= 16 && t < 32) {
      const int q = t - 16;
      s_b2[q] = (q < 4) ? enc_b2[q] : 0.0f;
    }
    if (t >= 32 && t < 36) {             // x_basis[q], q = t-32 (4 threads)
      const int q = t - 32;
      float hd[8];
#pragma unroll
      for (int p = 0; p < 8; ++p) {
        float s = dec_b1[p];
#pragma unroll
        for (int j = 0; j < 16; ++j) {
          const float basis = (float)(1u << j) * (1.0f / 65536.0f);
          s = fmaf(basis, dec_w1[p * 16 + j], s);
        }
        hd[p] = s > 0.0f ? s : 0.0f;
      }
      float z = dec_b2[q];
#pragma unroll
      for (int p = 0; p < 8; ++p) z = fmaf(hd[p], dec_w2[q * 8 + p], z);
      s_xb[q] = 1.0f / (1.0f + __expf(-z));  // sigmoid
    }
  }
  __syncthreads();

  const int lane = t & 31;
  const int wave = t >> 5;
  const int tile = blockIdx.x * WAVES_PER_BLOCK + wave;
  const int row0 = tile * ROWS_PER_WAVE;

  const int m    = lane & 15;   // row (A) or column (B/D) index within tile
  const int hi   = lane >> 4;   // 0 -> K/N pair {0,1}; 1 -> pair {2,3}
  const int koff = hi * 2;

  // ---- Quantization: nearest of 65536 affine levels (closed form).
  // levels[i] = (2i - 65535)/65536  =>  idx = round(x*32768 + 32767.5).
  const int row = row0 + m;
  const int rr  = (row < n) ? row : (n - 1);   // clamp: unconditional load
  const float xv = x[rr];
  const float fi = fmaf(xv, 32768.0f, 32767.5f);
  int idx = (int)floorf(fi + 0.5f);
  idx = idx < 0 ? 0 : (idx > 65535 ? 65535 : idx);

  // ---- Layer 1: bits(16x16) @ enc_w1^T(16x16 padded), 4 chunks of K=4.
  v8f acc = {};
#pragma unroll
  for (int kc = 0; kc < 4; ++kc) {
    const int k0 = kc * 4 + koff;
    v2f a, b;
    a.x = ((idx >> (k0 + 0)) & 1) ? 1.0f : -1.0f;   // A: M=m across lanes
    a.y = ((idx >> (k0 + 1)) & 1) ? 1.0f : -1.0f;
    b.x = s_w1t[k0 + 0][m];                         // B: N=m across lanes
    b.y = s_w1t[k0 + 1][m];
    acc = __builtin_amdgcn_wmma_f32_16x16x4_f32(
        false, a, false, b, (short)0, acc, false, false);
  }

  // D layout: VGPR v holds row M = v + 8*hi, col N = m. Bias + ReLU -> LDS.
  const float bias1 = s_b1[m];
#pragma unroll
  for (int v = 0; v < 8; ++v) {
    float h = acc[v] + bias1;
    lds_h1[wave][v + 8 * hi][m] = h > 0.0f ? h : 0.0f;
  }
  __syncthreads();

  // ---- Layer 2: h1(16x8) @ enc_w2^T(8x16 padded), 2 chunks of K=4.
  v8f acc2 = {};
#pragma unroll
  for (int kc = 0; kc < 2; ++kc) {
    const int k0 = kc * 4 + koff;
    v2f a, b;
    a.x = lds_h1[wave][m][k0 + 0];                  // transpose via LDS
    a.y = lds_h1[wave][m][k0 + 1];
    b.x = s_w2t[k0 + 0][m];
    b.y = s_w2t[k0 + 1][m];
    acc2 = __builtin_amdgcn_wmma_f32_16x16x4_f32(
        false, a, false, b, (short)0, acc2, false, false);
  }

  // sign(tanh(z)) == sign(z): no transcendental on the hot path.
  const float bias2 = s_b2[m];
  const float xb    = (m < 4) ? s_xb[m] : 0.0f;
#pragma unroll
  for (int v = 0; v < 8; ++v) {
    if (m < 4) {
      const float z = acc2[v] + bias2;
      const float sgn = (z > 0.0f) ? 1.0f : (z < 0.0f ? -1.0f : 0.0f);
      lds_code[wave][v + 8 * hi][m] = sgn * xb;
    }
  }
  __syncthreads();

  // ---- Final reduction over q (4 values) and store.
  if (lane < 16) {
    const int r = row0 + lane;
    if (r < n) {
      out[r] = lds_code[wave][lane][0] + lds_code[wave][lane][1] +
               lds_code[wave][lane][2] + lds_code[wave][lane][3];
    }
  }
}

extern "C" void kernel_launch(void* const* d_in, const int* in_sizes, int n_in,
                              void* d_out, int out_size, void* d_ws, size_t ws_size,
                              hipStream_t stream) {
  (void)n_in; (void)out_size; (void)d_ws; (void)ws_size;
  const float* xp      = (const float*)d_in[0];
  const float* enc_w1  = (const float*)d_in[1];
  const float* enc_b1  = (const float*)d_in[2];
  const float* enc_w2  = (const float*)d_in[3];
  const float* enc_b2  = (const float*)d_in[4];
  const float* dec_w1  = (const float*)d_in[5];
  const float* dec_b1  = (const float*)d_in[6];
  const float* dec_w2  = (const float*)d_in[7];
  const float* dec_b2  = (const float*)d_in[8];
  float* out = (float*)d_out;

  const int n = in_sizes[0];  // 4096 elements = 256 tiles of 16 rows
  const int rows_per_block = ROWS_PER_WAVE * WAVES_PER_BLOCK;  // 128
  const int blocks = (n + rows_per_block - 1) / rows_per_block;

  ae_wmma_kernel<<<blocks, 256, 0, stream>>>(
      xp, enc_w1, enc_b1, enc_w2, enc_b2,
      dec_w1, dec_b1, dec_w2, dec_b2, out, n);
}